// MaskedLinear_61452392071888
// MI455X (gfx1250) — compile-verified
//
#include <hip/hip_runtime.h>

typedef __attribute__((ext_vector_type(2))) float v2f;
typedef __attribute__((ext_vector_type(4))) float v4f;
typedef __attribute__((ext_vector_type(8))) float v8f;

// Problem constants (from reference): 64 modules of 64->64, batch 4096.
constexpr int NMOD        = 64;
constexpr int MF          = 64;            // features per module (in and out)
constexpr int D_IN        = NMOD * MF;     // 4096
constexpr int D_OUT       = NMOD * MF;     // 4096
constexpr int BATCH       = 4096;
constexpr int ROWS_PER_BLK = 32;           // 2 row-tiles x 4 col-tiles = 8 waves
constexpr int LDS_STRIDE  = 68;            // 64 + 4 pad -> conflict-free b64 reads

__global__ void __launch_bounds__(256)
masked_linear_wmma_f32(const float* __restrict__ x,
                       const float* __restrict__ W,
                       const float* __restrict__ bias,
                       float* __restrict__ y)
{
    __shared__ float ldsW[MF * LDS_STRIDE];   // W_m block, n-major, padded (17.4 KB)

    const int m    = blockIdx.y;              // module
    const int r0   = blockIdx.x * ROWS_PER_BLK;
    const int tid  = threadIdx.x;
    const int lane = tid & 31;                // wave32
    const int wid  = tid >> 5;                // 0..7

    // ---- Stage W_m (64x64 f32) into LDS, coalesced b128 loads ----
    // thread t: row n = t/4, 16 consecutive k at (t%4)*16
    {
        const int n = tid >> 2;
        const int c = (tid & 3) * 16;
        const float* src = W + (size_t)(m * MF + n) * D_IN + m * MF + c;
        float*       dst = &ldsW[n * LDS_STRIDE + c];
#pragma unroll
        for (int i = 0; i < 4; ++i) {
            v4f v = *(const v4f*)(src + i * 4);
            *(v4f*)(dst + i * 4) = v;         // 16B-aligned: 68*n and c are mult. of 4 floats
        }
    }
    __syncthreads();

    // ---- Per-wave 16x16 tile ----
    const int wrow = wid >> 2;                // 0..1  -> row tile
    const int ct   = wid & 3;                 // 0..3  -> col tile within module
    const int mr   = lane & 15;               // M for A, N for B/C/D
    const int koff = (lane >> 4) * 2;         // K sub-offset per ISA A/B f32 layout

    // A: x[r0 + wrow*16 + mr][m*64 + 4k + koff .. +1]  (global, b64, imm offsets)
    const float* ap = x + (size_t)(r0 + wrow * 16 + mr) * D_IN + m * MF + koff;
    // B: W_m^T fragment = ldsW[n][4k + koff .. +1], n = ct*16 + mr  (LDS, b64)
    const float* bp = &ldsW[(ct * 16 + mr) * LDS_STRIDE + koff];

    v8f acc = {0.f, 0.f, 0.f, 0.f, 0.f, 0.f, 0.f, 0.f};

#pragma unroll
    for (int kk = 0; kk < MF / 4; ++kk) {     // 16 x V_WMMA_F32_16X16X4_F32
        v2f a = *(const v2f*)(ap + kk * 4);
        v2f b = *(const v2f*)(bp + kk * 4);
        acc = __builtin_amdgcn_wmma_f32_16x16x4_f32(
            /*neg_a=*/false, a, /*neg_b=*/false, b,
            /*c_mod=*/(short)0, acc, /*reuse_a=*/false, /*reuse_b=*/false);
    }

    // ---- Bias (depends only on N = lane&15) + store ----
    const float bv = bias[m * MF + ct * 16 + mr];

    // D layout: VGPR r, lane L -> M = r + 8*(L>>4), N = L&15
    float* yp = y + (size_t)(r0 + wrow * 16 + 8 * (lane >> 4)) * D_OUT
                  + m * MF + ct * 16 + mr;
#pragma unroll
    for (int r = 0; r < 8; ++r) {
        yp[(size_t)r * D_OUT] = acc[r] + bv;
    }
}

extern "C" void kernel_launch(void* const* d_in, const int* in_sizes, int n_in,
                              void* d_out, int out_size, void* d_ws, size_t ws_size,
                              hipStream_t stream) {
    const float* x  = (const float*)d_in[0];
    const float* W  = (const float*)d_in[1];
    const float* b  = (const float*)d_in[2];
    // d_in[3] = num_of_modules (compile-time constant 64; ignored)
    float* y = (float*)d_out;

    dim3 grid(BATCH / ROWS_PER_BLK, NMOD);    // (128, 64)
    masked_linear_wmma_f32<<<grid, 256, 0, stream>>>(x, W, b, y);
}